// _Weight_17351667876255
// MI455X (gfx1250) — compile-verified
//
#include <hip/hip_runtime.h>

typedef __attribute__((ext_vector_type(2))) float v2f;
typedef __attribute__((ext_vector_type(8))) float v8f;

#define B_  8
#define N_  128
#define D_  64
#define DO_ 64

// ---- workspace layout (float offsets) ----
static constexpr size_t OFF_DIAG = 0;
static constexpr size_t OFF_ROW  = OFF_DIAG + (size_t)B_*N_*D_;
static constexpr size_t OFF_COL  = OFF_ROW  + (size_t)B_*N_*D_;
static constexpr size_t OFF_TR   = OFF_COL  + (size_t)B_*N_*D_;
static constexpr size_t OFF_OS   = OFF_TR   + (size_t)B_*D_;
static constexpr size_t OFF_WAT  = OFF_OS   + (size_t)B_*D_;        // Wa^T  [c*64+k]
static constexpr size_t OFF_WBT  = OFF_WAT  + (size_t)D_*DO_;       // Wb^T
static constexpr size_t OFF_MU   = OFF_WBT  + (size_t)D_*DO_;       // 192x64
static constexpr size_t OFF_MV   = OFF_MU   + (size_t)3*D_*DO_;
static constexpr size_t OFF_MD   = OFF_MV   + (size_t)3*D_*DO_;
static constexpr size_t OFF_U    = OFF_MD   + (size_t)3*D_*DO_;
static constexpr size_t OFF_V    = OFF_U    + (size_t)B_*N_*DO_;
static constexpr size_t OFF_DG   = OFF_V    + (size_t)B_*N_*DO_;
static constexpr size_t OFF_TV   = OFF_DG   + (size_t)B_*N_*DO_;
static constexpr size_t OFF_DC   = OFF_TV   + (size_t)B_*DO_;

__device__ __forceinline__ float mval(const unsigned char* m, int idx) {
    return m[idx] ? 1.0f : 0.0f;   // jnp bool -> 1 byte per element
}
__device__ __forceinline__ float wget(const float* w, int blk, int r, int c) {
    return w[((size_t)(blk * D_ + r)) * DO_ + c];   // weight (960,64) row-major
}

// ---- fuse weight blocks:  Wa^T, Wb^T, MU, MV, MD ----
__global__ void k_wprep(const float* __restrict__ w, float* __restrict__ ws) {
    int tid = blockIdx.x * blockDim.x + threadIdx.x;
    if (tid >= 3 * D_ * DO_) return;
    int k = tid / DO_;          // 0..191
    int c = tid % DO_;
    int kk = k & 63;
    float mu, mv, md;
    if (k < 64) {
        mu = wget(w,3,kk,c)  - wget(w,13,kk,c);        // W4 - W14
        mv = wget(w,4,kk,c)  - wget(w,13,kk,c);        // W5 - W14
        md = wget(w,0,kk,c)  - wget(w,11,kk,c);        // W1 - W12
        float wa = wget(w,1,kk,c) - wget(w,9,kk,c)  - wget(w,10,kk,c) + wget(w,14,kk,c); // W2-W10-W11+W15
        float wb = wget(w,2,kk,c) - wget(w,7,kk,c)  - wget(w,8,kk,c)  + wget(w,14,kk,c); // W3-W8-W9+W15
        ws[OFF_WAT + (size_t)c*64 + kk] = wa;          // transposed [col][k]
        ws[OFF_WBT + (size_t)c*64 + kk] = wb;
    } else if (k < 128) {
        mu = wget(w,10,kk,c) - wget(w,14,kk,c);        // W11 - W15  (row_i)
        mv = wget(w,7,kk,c)  - wget(w,14,kk,c);        // W8  - W15  (row_j)
        md = wget(w,5,kk,c)  - wget(w,12,kk,c);        // W6  - W13
    } else {
        mu = wget(w,8,kk,c)  - wget(w,14,kk,c);        // W9  - W15  (col_i)
        mv = wget(w,9,kk,c)  - wget(w,14,kk,c);        // W10 - W15  (col_j)
        md = wget(w,6,kk,c)  - wget(w,12,kk,c);        // W7  - W13
    }
    ws[OFF_MU + (size_t)k*DO_ + c] = mu;
    ws[OFF_MV + (size_t)k*DO_ + c] = mv;
    ws[OFF_MD + (size_t)k*DO_ + c] = md;
}

// ---- per (b,i): diagA, row, col (mask folded in) ----
__global__ void k_reduce(const float* __restrict__ A, const unsigned char* __restrict__ msk,
                         float* __restrict__ ws) {
    int bi = blockIdx.x;                // b*N_ + i
    int b = bi / N_, i = bi % N_;
    int d = threadIdx.x;                // 0..63
    float mi = mval(msk, bi);
    const float* Ab = A + (size_t)b * N_ * N_ * D_;
    float rs = 0.f, cs = 0.f;
    for (int j = 0; j < N_; ++j) {
        float mj = mval(msk, b * N_ + j);
        rs += Ab[((size_t)i * N_ + j) * D_ + d] * mj;
        cs += Ab[((size_t)j * N_ + i) * D_ + d] * mj;
    }
    float diag = Ab[((size_t)i * N_ + i) * D_ + d] * mi * mi;
    ws[OFF_DIAG + (size_t)bi * D_ + d] = diag;
    ws[OFF_ROW  + (size_t)bi * D_ + d] = mi * rs - diag;
    ws[OFF_COL  + (size_t)bi * D_ + d] = mi * cs - diag;
}

// ---- per b: trace, offsum ----
__global__ void k_batchsum(float* __restrict__ ws) {
    int b = blockIdx.x, d = threadIdx.x;
    float tr = 0.f, os = 0.f;
    for (int i = 0; i < N_; ++i) {
        tr += ws[OFF_DIAG + (size_t)(b * N_ + i) * D_ + d];
        os += ws[OFF_ROW  + (size_t)(b * N_ + i) * D_ + d];   // offsum = sum_i row_i
    }
    ws[OFF_TR + (size_t)b * D_ + d] = tr;
    ws[OFF_OS + (size_t)b * D_ + d] = os;
}

// ---- per b: T[b] = trace@W14 + offsum@W15 ; Dc[b] = trace@W12 + offsum@W13 ----
__global__ void k_bvec(const float* __restrict__ w, float* __restrict__ ws) {
    int b = blockIdx.x, c = threadIdx.x;
    float tv = 0.f, dc = 0.f;
    for (int k = 0; k < D_; ++k) {
        float tr = ws[OFF_TR + (size_t)b * D_ + k];
        float os = ws[OFF_OS + (size_t)b * D_ + k];
        tv += tr * wget(w,13,k,c) + os * wget(w,14,k,c);
        dc += tr * wget(w,11,k,c) + os * wget(w,12,k,c);
    }
    ws[OFF_TV + (size_t)b * DO_ + c] = tv;
    ws[OFF_DC + (size_t)b * DO_ + c] = dc;
}

// ---- per (b,i): U, V, DiagOut ----
__global__ void k_bias(float* __restrict__ ws) {
    __shared__ float X[3 * D_];
    int bi = blockIdx.x, b = bi / N_;
    int c = threadIdx.x;                // 0..63
    X[c]        = ws[OFF_DIAG + (size_t)bi * D_ + c];
    X[64 + c]   = ws[OFF_ROW  + (size_t)bi * D_ + c];
    X[128 + c]  = ws[OFF_COL  + (size_t)bi * D_ + c];
    __syncthreads();
    float u = 0.f, v = 0.f, dg = 0.f;
    for (int k = 0; k < 3 * D_; ++k) {
        float x = X[k];
        u  += x * ws[OFF_MU + (size_t)k * DO_ + c];
        v  += x * ws[OFF_MV + (size_t)k * DO_ + c];
        dg += x * ws[OFF_MD + (size_t)k * DO_ + c];
    }
    ws[OFF_U  + (size_t)bi * DO_ + c] = u;
    ws[OFF_V  + (size_t)bi * DO_ + c] = v;
    ws[OFF_DG + (size_t)bi * DO_ + c] = dg + ws[OFF_DC + (size_t)b * DO_ + c];
}

// ---- main WMMA kernel: one block per (b,i); 8 waves = 8 j-tiles of 16 ----
__global__ void __launch_bounds__(256) k_main(const float* __restrict__ A,
                                              const unsigned char* __restrict__ msk,
                                              const float* __restrict__ ws,
                                              float* __restrict__ out) {
    __shared__ float sWa[64 * 68];      // padded stride 68 -> conflict-free
    __shared__ float sWb[64 * 68];
    int bi = blockIdx.x;
    int b = bi / N_, i = bi % N_;

    for (int idx = threadIdx.x; idx < 64 * 64; idx += 256) {
        int c = idx >> 6, k = idx & 63;
        sWa[c * 68 + k] = ws[OFF_WAT + idx];
        sWb[c * 68 + k] = ws[OFF_WBT + idx];
    }
    __syncthreads();

    int wv    = threadIdx.x >> 5;       // wave id = M-tile (j-tile)
    int lane  = threadIdx.x & 31;
    int mrow  = lane & 15;              // A-frag: M = lane%16
    int khalf = lane >> 4;              // A-frag: K = khalf*2 + {0,1}
    int j0    = wv * 16 + mrow;         // this lane's j row

    float mi    = mval(msk, b * N_ + i);
    float scale = mi * mval(msk, b * N_ + j0);

    const float* r1 = A + ((size_t)(b * N_ + i)  * N_ + j0) * D_ + khalf * 2;  // Am[b,i,j0,:]
    const float* r2 = A + ((size_t)(b * N_ + j0) * N_ + i)  * D_ + khalf * 2;  // Am[b,j0,i,:]
    __builtin_prefetch(r1, 0, 0);
    __builtin_prefetch(r2, 0, 0);

    v2f a1[16], a2[16];
#pragma unroll
    for (int s = 0; s < 16; ++s) {      // K = 64 in 16 steps of 4
        v2f t1 = *(const v2f*)(r1 + s * 4);
        v2f t2 = *(const v2f*)(r2 + s * 4);
        a1[s] = t1 * scale;
        a2[s] = t2 * scale;
    }

    const float* Ub = ws + OFF_U  + (size_t)(b * N_ + i) * DO_;
    const float* Tb = ws + OFF_TV + (size_t)b * DO_;
    const float* Vb = ws + OFF_V  + (size_t)b * N_ * DO_;
    const float* Db = ws + OFF_DG + (size_t)(b * N_ + i) * DO_;
    float* outBase  = out + (size_t)(b * N_ + i) * N_ * DO_;

#pragma unroll
    for (int t = 0; t < 4; ++t) {       // 4 N-tiles over Dout=64
        v8f acc = {};
        const float* wa0 = &sWa[(t * 16 + mrow) * 68 + khalf * 2];
        const float* wb0 = &sWb[(t * 16 + mrow) * 68 + khalf * 2];
#pragma unroll
        for (int s = 0; s < 16; ++s) {
            v2f bwa = *(const v2f*)(wa0 + s * 4);
            acc = __builtin_amdgcn_wmma_f32_16x16x4_f32(false, a1[s], false, bwa,
                                                        (short)0, acc, false, false);
            v2f bwb = *(const v2f*)(wb0 + s * 4);
            acc = __builtin_amdgcn_wmma_f32_16x16x4_f32(false, a2[s], false, bwb,
                                                        (short)0, acc, false, false);
        }
        int col  = t * 16 + mrow;       // D-layout: N = lane%16
        float ub = Ub[col] + Tb[col];
#pragma unroll
        for (int v = 0; v < 8; ++v) {   // D-layout: M = v + khalf*8
            int jr = wv * 16 + v + khalf * 8;
            float val = acc[v] + ub + Vb[(size_t)jr * DO_ + col];
            if (jr == i) val = Db[col];               // diagonal override
            outBase[(size_t)jr * DO_ + col] = val;
        }
    }
}

extern "C" void kernel_launch(void* const* d_in, const int* in_sizes, int n_in,
                              void* d_out, int out_size, void* d_ws, size_t ws_size,
                              hipStream_t stream) {
    const float*         A   = (const float*)d_in[0];
    const unsigned char* msk = (const unsigned char*)d_in[1];   // bool, 1 byte/elem
    const float*         w   = (const float*)d_in[2];
    float* out = (float*)d_out;
    float* ws  = (float*)d_ws;

    k_wprep   <<<(3 * D_ * DO_ + 255) / 256, 256, 0, stream>>>(w, ws);
    k_reduce  <<<B_ * N_, D_,  0, stream>>>(A, msk, ws);
    k_batchsum<<<B_,      D_,  0, stream>>>(ws);
    k_bvec    <<<B_,      DO_, 0, stream>>>(w, ws);
    k_bias    <<<B_ * N_, DO_, 0, stream>>>(ws);
    k_main    <<<B_ * N_, 256, 0, stream>>>(A, msk, ws, out);
}